// TtMambaSSM_51891794871135
// MI455X (gfx1250) — compile-verified
//
#include <hip/hip_runtime.h>

typedef float v2f __attribute__((ext_vector_type(2)));
typedef float v4f __attribute__((ext_vector_type(4)));
typedef float v8f __attribute__((ext_vector_type(8)));

#define B_DIM 256
#define D_DIM 5120
#define N_DIM 16
#define R_DIM 160
#define KSPLIT 8

// workspace layout (float offsets)
#define PT1_OFF   0                                    // KSPLIT * 256*160
#define PBC_OFF   (PT1_OFF + KSPLIT * B_DIM * R_DIM)   // KSPLIT * 256*32
#define T1_OFF    (PBC_OFF + KSPLIT * B_DIM * 32)      // 256*160
#define BC_OFF    (T1_OFF + B_DIM * R_DIM)             // 256*32
#define ANEG_OFF  (BC_OFF + B_DIM * 32)                // 5120*16
#define BCDOT_OFF (ANEG_OFF + D_DIM * N_DIM)           // 256

// ---------------------------------------------------------------------------
// Kernel 1: split-K GEMM  x(256xD) @ W(DxLDB), one wave per 16x16 tile per
// K-split, fp32 WMMA 16x16x4. LDB is a template constant so all B-matrix
// strides fold into compile-time instruction offsets (no per-iter u64 math).
// ---------------------------------------------------------------------------
template <int LDB>
__global__ void __launch_bounds__(32)
gemm1_kernel(const float* __restrict__ x, const float* __restrict__ W,
             float* __restrict__ Pout) {
  const int lane = threadIdx.x & 31;
  const int half = lane >> 4;     // 0: lanes 0-15, 1: lanes 16-31
  const int l    = lane & 15;
  const int ntile = blockIdx.x;
  const int mtile = blockIdx.y;   // 0..15
  const int s     = blockIdx.z;   // 0..KSPLIT-1
  const int m0  = mtile * 16;
  const int col = ntile * 16 + l;

  const int kChunk = D_DIM / KSPLIT;        // 640, multiple of 16
  const int kBeg = s * kChunk;

  // ISA fp32 A layout: v0/v1 hold K = 2*half + {0,1}; B mirrored over lanes.
  const float* __restrict__ arow = x + (size_t)(m0 + l) * D_DIM + kBeg + 2 * half;
  const float* __restrict__ bcol = W + (size_t)(kBeg + 2 * half) * LDB + col;

  v8f acc = {};
#pragma unroll 4
  for (int k0 = 0; k0 < kChunk; k0 += 4) {
    v2f av, bv;
    av.x = arow[k0];
    av.y = arow[k0 + 1];
    bv.x = bcol[(size_t)k0 * LDB];
    bv.y = bcol[(size_t)k0 * LDB + LDB];
    acc = __builtin_amdgcn_wmma_f32_16x16x4_f32(false, av, false, bv,
                                                (short)0, acc, false, false);
  }

  float* __restrict__ P = Pout + (size_t)s * (B_DIM * LDB);
#pragma unroll
  for (int i = 0; i < 8; ++i) {             // C: VGPR i -> row m0 + 8*half + i
    P[(size_t)(m0 + half * 8 + i) * LDB + col] = acc[i];
  }
}

// ---------------------------------------------------------------------------
// Kernel 2: sum split-K partials -> T1 (256x160) and BC (256x32)
// ---------------------------------------------------------------------------
__global__ void __launch_bounds__(256)
reduce_kernel(const float* __restrict__ Pt1, const float* __restrict__ Pbc,
              float* __restrict__ T1, float* __restrict__ BC) {
  const int idx = blockIdx.x * 256 + threadIdx.x;   // 0 .. 49151
  if (idx < B_DIM * R_DIM) {
    float s = 0.f;
#pragma unroll
    for (int j = 0; j < KSPLIT; ++j) s += Pt1[(size_t)j * (B_DIM * R_DIM) + idx];
    T1[idx] = s;
  } else {
    const int k = idx - B_DIM * R_DIM;              // 0 .. 8191
    float s = 0.f;
#pragma unroll
    for (int j = 0; j < KSPLIT; ++j) s += Pbc[(size_t)j * (B_DIM * 32) + k];
    BC[k] = s;
  }
}

// ---------------------------------------------------------------------------
// Kernel 3: Aneg = -exp(A_log)   (5120*16 elements, computed once)
// ---------------------------------------------------------------------------
__global__ void __launch_bounds__(256)
aneg_kernel(const float* __restrict__ Alog, float* __restrict__ Aneg) {
  const int idx = blockIdx.x * 256 + threadIdx.x;   // 0 .. 81919
  Aneg[idx] = -__expf(Alog[idx]);
}

// ---------------------------------------------------------------------------
// Kernel 4: bcdot[b] = sum_n Bm[b,n]*Cm[b,n]
// ---------------------------------------------------------------------------
__global__ void __launch_bounds__(256)
bcdot_kernel(const float* __restrict__ BC, float* __restrict__ bcdot) {
  const int b = threadIdx.x;                        // 256 threads, 1 block
  const float* __restrict__ row = BC + b * 32;
  float s = 0.f;
#pragma unroll
  for (int n = 0; n < N_DIM; ++n) s = fmaf(row[n], row[16 + n], s);
  bcdot[b] = s;
}

// ---------------------------------------------------------------------------
// Kernel 5: dt = softplus(T1 @ W_dt + b_dt)  -> written into d_out (256x5120)
// ---------------------------------------------------------------------------
__global__ void __launch_bounds__(32)
gemm2_kernel(const float* __restrict__ T1, const float* __restrict__ Wdt,
             const float* __restrict__ bdt, float* __restrict__ dt_out) {
  const int lane = threadIdx.x & 31;
  const int half = lane >> 4;
  const int l    = lane & 15;
  const int ntile = blockIdx.x;   // 0..319
  const int mtile = blockIdx.y;   // 0..15
  const int m0  = mtile * 16;
  const int col = ntile * 16 + l;

  const float* __restrict__ arow = T1 + (size_t)(m0 + l) * R_DIM + 2 * half;
  const float* __restrict__ bcol = Wdt + (size_t)(2 * half) * D_DIM + col;

  v8f acc = {};
#pragma unroll 4
  for (int k0 = 0; k0 < R_DIM; k0 += 4) {
    v2f av, bv;
    av.x = arow[k0];
    av.y = arow[k0 + 1];
    bv.x = bcol[(size_t)k0 * D_DIM];
    bv.y = bcol[(size_t)k0 * D_DIM + D_DIM];
    acc = __builtin_amdgcn_wmma_f32_16x16x4_f32(false, av, false, bv,
                                                (short)0, acc, false, false);
  }

  const float bias = bdt[col];
#pragma unroll
  for (int i = 0; i < 8; ++i) {
    const float v = acc[i] + bias;
    // numerically stable softplus
    const float sp = fmaxf(v, 0.f) + log1pf(__expf(-fabsf(v)));
    dt_out[(size_t)(m0 + half * 8 + i) * D_DIM + col] = sp;
  }
}

// ---------------------------------------------------------------------------
// Kernel 6 (bandwidth-critical): stream h0 (84 MB, read-once) with
// non-temporal loads; small reused tensors (x, BC, Aneg) stay cache-resident.
// y[b,d] = sum_n exp(Aneg[d,n]*dt)*h0[b,d,n]*Cm[b,n] + dt*x*bcdot[b] + x
// dt is read from d_out and overwritten with y (same thread, no hazard).
// ---------------------------------------------------------------------------
__global__ void __launch_bounds__(256)
final_kernel(const float* __restrict__ x, const float* __restrict__ h0,
             const float* __restrict__ Aneg, const float* __restrict__ BC,
             const float* __restrict__ bcdot, float* __restrict__ y) {
  const int d = blockIdx.x * 256 + threadIdx.x;     // D/256 = 20 blocks
  const int b = blockIdx.y;                         // 256
  const size_t bd = (size_t)b * D_DIM + d;

  const float dt = y[bd];
  const float xv = x[bd];
  const v4f* __restrict__ hp = (const v4f*)(h0 + bd * N_DIM);         // 64B aligned
  const v4f* __restrict__ ap = (const v4f*)(Aneg + (size_t)d * N_DIM);
  const float* __restrict__ Cm = BC + b * 32 + 16;

  float acc = 0.f;
#pragma unroll
  for (int q = 0; q < 4; ++q) {
    const v4f hv = __builtin_nontemporal_load(hp + q);   // TH=NT: read-once stream
    const v4f av = ap[q];
#pragma unroll
    for (int j = 0; j < 4; ++j) {
      const float da = __expf(av[j] * dt);
      acc = fmaf(da * hv[j], Cm[q * 4 + j], acc);
    }
  }
  __builtin_nontemporal_store(acc + dt * xv * bcdot[b] + xv, &y[bd]);
}

// ---------------------------------------------------------------------------
extern "C" void kernel_launch(void* const* d_in, const int* in_sizes, int n_in,
                              void* d_out, int out_size, void* d_ws, size_t ws_size,
                              hipStream_t stream) {
  const float* x    = (const float*)d_in[0];
  const float* h0   = (const float*)d_in[1];
  const float* Wxdt = (const float*)d_in[2];
  const float* Wdt  = (const float*)d_in[3];
  const float* bdt  = (const float*)d_in[4];
  const float* Wbc  = (const float*)d_in[5];
  const float* Alog = (const float*)d_in[6];

  float* ws    = (float*)d_ws;
  float* Pt1   = ws + PT1_OFF;
  float* Pbc   = ws + PBC_OFF;
  float* T1    = ws + T1_OFF;
  float* BC    = ws + BC_OFF;
  float* Aneg  = ws + ANEG_OFF;
  float* Bcdot = ws + BCDOT_OFF;
  float* y     = (float*)d_out;

  gemm1_kernel<R_DIM><<<dim3(10, 16, KSPLIT), 32, 0, stream>>>(x, Wxdt, Pt1);
  gemm1_kernel<32>   <<<dim3(2, 16, KSPLIT), 32, 0, stream>>>(x, Wbc, Pbc);
  reduce_kernel<<<dim3(192), 256, 0, stream>>>(Pt1, Pbc, T1, BC);
  aneg_kernel  <<<dim3(320), 256, 0, stream>>>(Alog, Aneg);
  bcdot_kernel <<<dim3(1), 256, 0, stream>>>(BC, Bcdot);
  gemm2_kernel <<<dim3(320, 16), 32, 0, stream>>>(T1, Wdt, bdt, y);
  final_kernel <<<dim3(20, 256), 256, 0, stream>>>(x, h0, Aneg, BC, Bcdot, y);
}